// SE3_Quadratic_24996709663344
// MI455X (gfx1250) — compile-verified
//
#include <hip/hip_runtime.h>
#include <hip/hip_bf16.h>
#include <math.h>

typedef __attribute__((ext_vector_type(2))) float v2f;
typedef __attribute__((ext_vector_type(8))) float v8f;

// wave32 lane gather: Dst[lane] = src[srcLane]  (full-EXEC use only)
__device__ __forceinline__ float shfl32(float v, int srcLane) {
  return __int_as_float(__builtin_amdgcn_ds_bpermute(srcLane << 2, __float_as_int(v)));
}

__device__ __forceinline__ void mm3(const float* __restrict__ A,
                                    const float* __restrict__ B,
                                    float* __restrict__ C) {
#pragma unroll
  for (int i = 0; i < 3; ++i)
#pragma unroll
    for (int j = 0; j < 3; ++j)
      C[i * 3 + j] = fmaf(A[i * 3 + 0], B[0 * 3 + j],
                     fmaf(A[i * 3 + 1], B[1 * 3 + j], A[i * 3 + 2] * B[2 * 3 + j]));
}

__device__ __forceinline__ void mv3(const float* __restrict__ A,
                                    const float* __restrict__ v,
                                    float* __restrict__ r) {
#pragma unroll
  for (int i = 0; i < 3; ++i)
    r[i] = fmaf(A[i * 3 + 0], v[0], fmaf(A[i * 3 + 1], v[1], A[i * 3 + 2] * v[2]));
}

__device__ __forceinline__ void skew3(const float* __restrict__ w, float* __restrict__ S) {
  S[0] = 0.0f;  S[1] = -w[2]; S[2] = w[1];
  S[3] = w[2];  S[4] = 0.0f;  S[5] = -w[0];
  S[6] = -w[1]; S[7] = w[0];  S[8] = 0.0f;
}

__global__ __launch_bounds__(256) void se3_quadratic_kernel(
    const float* __restrict__ x,
    const float* __restrict__ I_diag,
    const float* __restrict__ Kd,
    const float* __restrict__ Go,
    const float* __restrict__ Gt,
    const float* __restrict__ Hd,
    float* __restrict__ out,
    int Btot)
{
  const int  lane = (int)(threadIdx.x & 31u);
  const bool loH  = lane < 16;
  const int  eRaw = (int)(blockIdx.x * blockDim.x + threadIdx.x);
  const int  e    = (eRaw < Btot) ? eRaw : (Btot - 1);   // clamp: keep EXEC full for WMMA

  // ---------------- per-element state ----------------
  const float* __restrict__ xr = x + (size_t)e * 13;
  float qi[6], P[6], T[6];
#pragma unroll
  for (int i = 0; i < 6; ++i) qi[i] = xr[i];
#pragma unroll
  for (int i = 0; i < 6; ++i) P[i] = xr[6 + i];
#pragma unroll
  for (int i = 0; i < 6; ++i) T[i] = P[i] / I_diag[i];

  // ============ yKd = Kd @ T  via V_WMMA_F32_16X16X4_F32 (exact fp32) ============
  // D(16x16) = A(16x4) x B(4x16) + C.
  // A rows m = Kd rows (rows 6..15 zero). B cols n = 16 consecutive batch elements.
  // K = 6 handled as two K-steps (k0 = 0..3, k1 = 4..7 with 6,7 zero padded).
  // Tile 0 covers this wave's lanes 0..15, tile 1 covers lanes 16..31.
  const int s0 = lane & 15;
  const int s1 = s0 + 16;

  // All shuffles unconditional / straight-line (full EXEC).
  float T0a = shfl32(T[0], s0), T1a = shfl32(T[1], s0), T2a = shfl32(T[2], s0);
  float T3a = shfl32(T[3], s0), T4a = shfl32(T[4], s0), T5a = shfl32(T[5], s0);
  float T0b = shfl32(T[0], s1), T1b = shfl32(T[1], s1), T2b = shfl32(T[2], s1);
  float T3b = shfl32(T[3], s1), T4b = shfl32(T[4], s1), T5b = shfl32(T[5], s1);

  // B fragment layout (4x16 f32): VGPR0 = K0 (lanes 0-15) / K2 (lanes 16-31),
  //                               VGPR1 = K1 / K3.
  v2f b_k0_t0, b_k1_t0, b_k0_t1, b_k1_t1;
  b_k0_t0[0] = loH ? T0a : T2a;  b_k0_t0[1] = loH ? T1a : T3a;
  b_k1_t0[0] = loH ? T4a : 0.0f; b_k1_t0[1] = loH ? T5a : 0.0f;
  b_k0_t1[0] = loH ? T0b : T2b;  b_k0_t1[1] = loH ? T1b : T3b;
  b_k1_t1[0] = loH ? T4b : 0.0f; b_k1_t1[1] = loH ? T5b : 0.0f;

  // A fragment layout (16x4 f32): VGPR0 = K0 (lanes 0-15) / K2 (lanes 16-31),
  //                               VGPR1 = K1 / K3.  Row = lane&15.
  const int r = s0;
  float a00 = 0.0f, a01 = 0.0f, a10 = 0.0f, a11 = 0.0f;
  if (r < 6) {
    a00 = Kd[r * 6 + (loH ? 0 : 2)];
    a01 = Kd[r * 6 + (loH ? 1 : 3)];
    a10 = loH ? Kd[r * 6 + 4] : 0.0f;   // K=6 column is zero pad
    a11 = loH ? Kd[r * 6 + 5] : 0.0f;   // K=7 column is zero pad
  }
  v2f a_k0, a_k1;
  a_k0[0] = a00; a_k0[1] = a01;
  a_k1[0] = a10; a_k1[1] = a11;

  v8f acc0 = {0.f, 0.f, 0.f, 0.f, 0.f, 0.f, 0.f, 0.f};
  v8f acc1 = {0.f, 0.f, 0.f, 0.f, 0.f, 0.f, 0.f, 0.f};
  acc0 = __builtin_amdgcn_wmma_f32_16x16x4_f32(false, a_k0, false, b_k0_t0, (short)0, acc0, false, false);
  acc0 = __builtin_amdgcn_wmma_f32_16x16x4_f32(false, a_k1, false, b_k1_t0, (short)0, acc0, false, false);
  acc1 = __builtin_amdgcn_wmma_f32_16x16x4_f32(false, a_k0, false, b_k0_t1, (short)0, acc1, false, false);
  acc1 = __builtin_amdgcn_wmma_f32_16x16x4_f32(false, a_k1, false, b_k1_t1, (short)0, acc1, false, false);

  // D: VGPR j, lane n (n<16) = (Kd @ T_{tileBase+n})[j].  Redistribute tile1 to high lanes.
  float yKd[6];
#pragma unroll
  for (int j = 0; j < 6; ++j) {
    float own = acc0[j];
    float sh  = shfl32(acc1[j], s0);
    yKd[j] = loH ? own : sh;
  }

  // ---------------- exp_SE3(qi) ----------------
  float w[3]  = {qi[0], qi[1], qi[2]};
  float vv[3] = {qi[3], qi[4], qi[5]};
  float th2 = w[0] * w[0] + w[1] * w[1] + w[2] * w[2] + 1e-30f;
  float th  = sqrtf(th2);
  bool  sm  = th < 1e-4f;
  float ths = sm ? 1.0f : th;
  float sn, cs;
  sincosf(ths, &sn, &cs);
  float Ac = sm ? 1.0f - th2 * (1.0f / 6.0f)   : sn / ths;
  float Bc = sm ? 0.5f - th2 * (1.0f / 24.0f)  : (1.0f - cs) / (ths * ths);
  float Cc = sm ? (1.0f / 6.0f) - th2 * (1.0f / 120.0f) : (ths - sn) / (ths * ths * ths);

  float W[9], W2[9];
  skew3(w, W);
  mm3(W, W, W2);
  float R[9], Vm[9];
#pragma unroll
  for (int i = 0; i < 9; ++i) { R[i] = Ac * W[i] + Bc * W2[i]; Vm[i] = Bc * W[i] + Cc * W2[i]; }
  R[0] += 1.0f; R[4] += 1.0f; R[8] += 1.0f;
  Vm[0] += 1.0f; Vm[4] += 1.0f; Vm[8] += 1.0f;
  float p[3];
  mv3(Vm, vv, p);

  // ---------------- H_D_CS = inv(H_I_D) @ H_I_B ----------------
  float Rdt[9], pd[3];
#pragma unroll
  for (int i = 0; i < 3; ++i) {
#pragma unroll
    for (int j = 0; j < 3; ++j) Rdt[j * 3 + i] = Hd[i * 4 + j];  // Rd^T
    pd[i] = Hd[i * 4 + 3];
  }
  float Rc[9];
  mm3(Rdt, R, Rc);
  float pmd[3] = {p[0] - pd[0], p[1] - pd[1], p[2] - pd[2]};
  float xc[3];
  mv3(Rdt, pmd, xc);

  // ---------------- log_SE3(H_I_B) ----------------
  float tr  = R[0] + R[4] + R[8];
  float cc  = fminf(fmaxf((tr - 1.0f) * 0.5f, -1.0f + 1e-7f), 1.0f - 1e-7f);
  float thl = acosf(cc);
  bool  sml = thl < 1e-4f;
  float thsl = sml ? 1.0f : thl;
  float snl, csl;
  sincosf(thsl, &snl, &csl);
  float al = sml ? 0.5f + thl * thl * (1.0f / 12.0f) : thsl / (2.0f * snl);
  float Wl[9];
#pragma unroll
  for (int i = 0; i < 3; ++i)
#pragma unroll
    for (int j = 0; j < 3; ++j) Wl[i * 3 + j] = al * (R[i * 3 + j] - R[j * 3 + i]);
  float Wl2[9];
  mm3(Wl, Wl, Wl2);
  float Dl = sml ? (1.0f / 12.0f)
                 : 1.0f / (thsl * thsl) - (1.0f + csl) / (2.0f * thsl * snl);
  float Vinv[9];
#pragma unroll
  for (int i = 0; i < 9; ++i) Vinv[i] = -0.5f * Wl[i] + Dl * Wl2[i];
  Vinv[0] += 1.0f; Vinv[4] += 1.0f; Vinv[8] += 1.0f;
  float vlog[3];
  mv3(Vinv, p, vlog);
  float wq[3] = {Wl[7], Wl[2], Wl[3]};  // unskew

  // ---------------- Jinv_SE3(qlog) ----------------
  float th2j = wq[0] * wq[0] + wq[1] * wq[1] + wq[2] * wq[2] + 1e-30f;
  float thj  = sqrtf(th2j);
  bool  smj  = thj < 1e-4f;
  float thsj = smj ? 1.0f : thj;
  float snj, csj;
  sincosf(thsj, &snj, &csj);
  float t2 = thsj * thsj, t3 = t2 * thsj, t4 = t3 * thsj, t5 = t4 * thsj;
  float Dj = smj ? (1.0f / 12.0f) : 1.0f / t2 - (1.0f + csj) / (2.0f * thsj * snj);
  float c1 = smj ? (1.0f / 6.0f)   : (thsj - snj) / t3;
  float c2 = smj ? (-1.0f / 24.0f) : (1.0f - 0.5f * t2 - csj) / t4;
  float c3 = smj ? (-1.0f / 120.0f): (thsj - snj - t3 * (1.0f / 6.0f)) / t5;

  float Wj[9], Vj[9], Wj2[9];
  skew3(wq, Wj);
  skew3(vlog, Vj);
  mm3(Wj, Wj, Wj2);
  float Jv[9];
#pragma unroll
  for (int i = 0; i < 9; ++i) Jv[i] = -0.5f * Wj[i] + Dj * Wj2[i];
  Jv[0] += 1.0f; Jv[4] += 1.0f; Jv[8] += 1.0f;

  float WV[9], VW[9], WVW[9], W2V[9], VW2[9], WVW2[9], W2VW[9];
  mm3(Wj, Vj, WV);
  mm3(Vj, Wj, VW);
  mm3(WV, Wj, WVW);
  mm3(Wj2, Vj, W2V);
  mm3(VW, Wj, VW2);
  mm3(WVW, Wj, WVW2);
  mm3(Wj, WVW, W2VW);
  float cq = -0.5f * (c2 - 3.0f * c3);
  float Q[9];
#pragma unroll
  for (int i = 0; i < 9; ++i)
    Q[i] = 0.5f * Vj[i] + c1 * (WV[i] + VW[i] + WVW[i])
         + c2 * (W2V[i] + VW2[i] - 3.0f * WVW[i]) + cq * (WVW2[i] + W2VW[i]);
  float JQ[9], QJ[9];
  mm3(Jv, Q, JQ);
  mm3(JQ, Jv, QJ);
#pragma unroll
  for (int i = 0; i < 9; ++i) QJ[i] = -QJ[i];

  float Tt[3] = {T[0], T[1], T[2]}, Tv[3] = {T[3], T[4], T[5]};
  float dq0[3], dq1a[3], dq1b[3];
  mv3(Jv, Tt, dq0);
  mv3(QJ, Tt, dq1a);
  mv3(Jv, Tv, dq1b);

  // ---------------- ad(T)^T @ P ----------------
  float Wt[9], Vt[9];
  skew3(Tt, Wt);
  skew3(Tv, Vt);
  float Pt[3] = {P[0], P[1], P[2]}, Pv[3] = {P[3], P[4], P[5]};
  float u1[3], u2[3], u3[3];
  mv3(Wt, Pt, u1);
  mv3(Vt, Pv, u2);
  mv3(Wt, Pv, u3);

  // ---------------- W_p(H_D_CS, Go, Gt) ----------------
  float GoM[9], GtM[9];
#pragma unroll
  for (int i = 0; i < 9; ++i) { GoM[i] = Go[i]; GtM[i] = Gt[i]; }
  float Rct[9];
#pragma unroll
  for (int i = 0; i < 3; ++i)
#pragma unroll
    for (int j = 0; j < 3; ++j) Rct[i * 3 + j] = Rc[j * 3 + i];
  float Sx[9], Sx2[9];
  skew3(xc, Sx);
  mm3(Sx, Sx, Sx2);
  float GoR[9];
  mm3(GoM, Rc, GoR);
  float RtS2[9], RtS2R[9], M2[9];
  mm3(Rct, Sx2, RtS2);
  mm3(RtS2, Rc, RtS2R);
  mm3(GtM, RtS2R, M2);
  float wp0 = -(GoR[7] - GoR[5]) - 0.5f * (M2[7] - M2[5]);
  float wp1 = -(GoR[2] - GoR[6]) - 0.5f * (M2[2] - M2[6]);
  float wp2 = -(GoR[3] - GoR[1]) - 0.5f * (M2[3] - M2[1]);

  float GtS[9], skGtS[9];
  mm3(GtM, Sx, GtS);
#pragma unroll
  for (int i = 0; i < 3; ++i)
#pragma unroll
    for (int j = 0; j < 3; ++j) skGtS[i * 3 + j] = 0.5f * (GtS[i * 3 + j] - GtS[j * 3 + i]);
  float tmpA[9], F1[9];
  mm3(Rct, skGtS, tmpA);
  mm3(tmpA, Rc, F1);
  float RtS[9], RtSR[9], M3[9];
  mm3(Rct, Sx, RtS);
  mm3(RtS, Rc, RtSR);
  mm3(GtM, RtSR, M3);
  float wp3 = -F1[7] - 0.5f * (M3[7] - M3[5]);
  float wp4 = -F1[2] - 0.5f * (M3[2] - M3[6]);
  float wp5 = -F1[3] - 0.5f * (M3[3] - M3[1]);

  // ---------------- combine + store ----------------
  if (eRaw < Btot) {
    float* __restrict__ o = out + (size_t)eRaw * 13;
    o[0]  = dq0[0];
    o[1]  = dq0[1];
    o[2]  = dq0[2];
    o[3]  = dq1a[0] + dq1b[0];
    o[4]  = dq1a[1] + dq1b[1];
    o[5]  = dq1a[2] + dq1b[2];
    o[6]  = -(u1[0] + u2[0]) - yKd[0] + wp0;
    o[7]  = -(u1[1] + u2[1]) - yKd[1] + wp1;
    o[8]  = -(u1[2] + u2[2]) - yKd[2] + wp2;
    o[9]  = -u3[0] - yKd[3] + wp3;
    o[10] = -u3[1] - yKd[4] + wp4;
    o[11] = -u3[2] - yKd[5] + wp5;
    o[12] = 0.0f;
  }
}

extern "C" void kernel_launch(void* const* d_in, const int* in_sizes, int n_in,
                              void* d_out, int out_size, void* d_ws, size_t ws_size,
                              hipStream_t stream) {
  (void)n_in; (void)out_size; (void)d_ws; (void)ws_size;
  // inputs: t(1), x(B*13), I_diag(6), Kd(36), Go(9), Gt(9), H_I_D(16)
  const float* x      = (const float*)d_in[1];
  const float* I_diag = (const float*)d_in[2];
  const float* Kd     = (const float*)d_in[3];
  const float* Go     = (const float*)d_in[4];
  const float* Gt     = (const float*)d_in[5];
  const float* Hd     = (const float*)d_in[6];
  float* out = (float*)d_out;

  const int B = in_sizes[1] / 13;
  const int threads = 256;
  const int blocks = (B + threads - 1) / threads;
  se3_quadratic_kernel<<<blocks, threads, 0, stream>>>(x, I_diag, Kd, Go, Gt, Hd, out, B);
}